// StructNConv2D_gx_with_ds_16320875725218
// MI455X (gfx1250) — compile-verified
//
#include <hip/hip_runtime.h>

// ---------------------------------------------------------------------------
// StructNConv2D_gx_with_ds for MI455X (gfx1250, wave32)
// Stage1+2+3 fused streaming kernel (memory-bound, ~290MB traffic)
// Stage4 channel mix via V_WMMA_F32_16X16X4_F32 (fp32 GEMM, 65536x32x32 x2)
// ---------------------------------------------------------------------------

#define EPSF 1e-20f

// problem constants
#define Bn  4
#define Cn  32
#define On  32
#define Hn  256
#define Wn  256
#define HOn 128
#define WOn 128
#define Pn  65536       // Bn*HOn*WOn
#define HWo 16384       // HOn*WOn

typedef __attribute__((ext_vector_type(2))) float v2f;
typedef __attribute__((ext_vector_type(8))) float v8f;

__device__ __forceinline__ float softplusf(float x) {
    // jax.nn.softplus = log1p(exp(x)), stable for large x
    return (x > 20.f) ? x : log1pf(expf(x));
}

// ---------------------------------------------------------------------------
// Kernel A: positive reparametrization of weights + global sums (1 block)
//   wp[32], swp[288], cwp[1024], ssw[1], scw[1]
// ---------------------------------------------------------------------------
__global__ __launch_bounds__(256) void setup_kernel(
    const float* __restrict__ wprop, const float* __restrict__ spw,
    const float* __restrict__ chw,
    float* __restrict__ wp, float* __restrict__ swp, float* __restrict__ cwp,
    float* __restrict__ ssw, float* __restrict__ scw)
{
    __shared__ float red[256];
    const int tid = threadIdx.x;

    float s1 = 0.f, s2 = 0.f;
    for (int i = tid; i < Cn; i += 256) wp[i] = softplusf(wprop[i]);
    for (int i = tid; i < Cn * 9; i += 256) { float v = softplusf(spw[i]); swp[i] = v; s1 += v; }
    for (int i = tid; i < On * Cn; i += 256) { float v = softplusf(chw[i]); cwp[i] = v; s2 += v; }

    red[tid] = s1;
    __syncthreads();
    for (int st = 128; st > 0; st >>= 1) {
        if (tid < st) red[tid] += red[tid + st];
        __syncthreads();
    }
    if (tid == 0) ssw[0] = red[0];
    __syncthreads();

    red[tid] = s2;
    __syncthreads();
    for (int st = 128; st > 0; st >>= 1) {
        if (tid < st) red[tid] += red[tid + st];
        __syncthreads();
    }
    if (tid == 0) scw[0] = red[0];
}

// ---------------------------------------------------------------------------
// Kernel B: fused stage1 (masked-edge elementwise) + 9-tap strided gather +
// per-channel spatial reduce.  One thread per output element (b,c,ho,wo).
// Writes t0 = cgx_sp*gx_sp and t1 = cgx_sp in channel-major [C][P] layout
// so the WMMA stage reads contiguous B-rows.
// ---------------------------------------------------------------------------
__global__ __launch_bounds__(256) void fused_spatial_kernel(
    const float* __restrict__ d,  const float* __restrict__ cd,
    const float* __restrict__ s,  const float* __restrict__ gx,
    const float* __restrict__ cgx, const float* __restrict__ sprod,
    const float* __restrict__ wp, const float* __restrict__ swp,
    const float* __restrict__ ssw,
    float* __restrict__ t0, float* __restrict__ t1)
{
    const int idx = blockIdx.x * 256 + threadIdx.x;   // [0, Bn*Cn*HOn*WOn)
    const int wo = idx & (WOn - 1);
    const int ho = (idx >> 7) & (HOn - 1);
    const int c  = (idx >> 14) & (Cn - 1);
    const int b  = idx >> 19;

    const float wpv  = wp[c];
    const float S_sw = ssw[0];
    const int base_in = ((b << 5) + c) << 16;               // (b*C+c)*H*W
    const int sp_base = ((b << 5) + c) * 9 * HWo + (ho << 7) + wo;

    float nom = 0.f, den = 0.f;
#pragma unroll
    for (int k = 0; k < 9; ++k) {
        const int i = k / 3, j = k % 3;
        const int h = 2 * ho - 1 + i;
        const int w = 2 * wo - 1 + j;
        if (h >= 0 && h < Hn && w >= 0 && w < Wn) {
            const int off = base_in + (h << 8) + w;
            const float dv = d[off],  cdv = cd[off], sv = s[off];
            const float gxv = gx[off], cgxv = cgx[off];

            // reference pad_r/pad_l == edge-column masking (not a shift)
            const float mL = (w < Wn - 1) ? 1.f : 0.f;
            const float mR = (w > 0)      ? 1.f : 0.f;
            const float dl = dv * mL,  dr = dv * mR;
            const float cdl = cdv * mL, cdr = cdv * mR;
            const float sl = sv * mL,  sr = sv * mR;

            const float cgx_ds = sv * sl * sr * cdl * cdr;
            const float height = (cdl * dl + cdr * dr) / (cdl + cdr + EPSF);
            const float gx_ds  = (dr - dl) * 0.5f / (height + EPSF);
            const float gxn  = (wpv * cgxv * gxv + cgx_ds * gx_ds) /
                               (wpv * cgxv + cgx_ds + EPSF);
            const float cgxn = (wpv * cgxv + cgx_ds) / (wpv + 1.f);

            const float spv = sprod[sp_base + k * HWo];
            const float swv = swp[c * 9 + k];
            const float cgp = cgxn * spv;
            nom += cgp * gxn * swv;
            den += cgp * swv;
        }
    }
    const float gx_sp  = nom / (den + EPSF);
    const float cgx_sp = den / (S_sw + EPSF);

    const int p = (b << 14) | (ho << 7) | wo;   // pixel index in [0,Pn)
    t0[(c << 16) + p] = cgx_sp * gx_sp;
    t1[(c << 16) + p] = cgx_sp;
}

// ---------------------------------------------------------------------------
// Kernel C: channel stage as WMMA f32 GEMM.
// Per wave: 16 pixels x 32 outputs, 4 accumulators (t0/t1 x O-tiles 0/1),
// K = C = 32 consumed in 8 steps of V_WMMA_F32_16X16X4_F32.
// A 16x4 layout: lane holds M = lane&15; VGPR0 K = k0+2*half, VGPR1 K+1.
// B 4x16 layout mirrors it with N = lane&15.  D: M = v + 8*half, N = lane&15.
// ---------------------------------------------------------------------------
__global__ __launch_bounds__(256) void channel_wmma_kernel(
    const float* __restrict__ t0, const float* __restrict__ t1,
    const float* __restrict__ cwp, const float* __restrict__ scw,
    const float* __restrict__ bias, float* __restrict__ out)
{
    const int wave = (blockIdx.x * 256 + threadIdx.x) >> 5;  // [0, 4096)
    const int lane = threadIdx.x & 31;
    const int n    = lane & 15;
    const int half = lane >> 4;
    const int p    = (wave << 4) + n;     // this lane's pixel column
    const float S_cw = scw[0];

    v8f acc00 = {}, acc01 = {}, acc10 = {}, acc11 = {};
#pragma unroll
    for (int k0 = 0; k0 < Cn; k0 += 4) {
        const int ka = k0 + (half << 1);
        const int kb = ka + 1;
        v2f b0, b1, a0, a1;
        b0.x = t0[ka * Pn + p];  b0.y = t0[kb * Pn + p];
        b1.x = t1[ka * Pn + p];  b1.y = t1[kb * Pn + p];
        a0.x = cwp[n * Cn + ka];         a0.y = cwp[n * Cn + kb];
        a1.x = cwp[(16 + n) * Cn + ka];  a1.y = cwp[(16 + n) * Cn + kb];
        acc00 = __builtin_amdgcn_wmma_f32_16x16x4_f32(false, a0, false, b0,
                                                      (short)0, acc00, false, false);
        acc01 = __builtin_amdgcn_wmma_f32_16x16x4_f32(false, a0, false, b1,
                                                      (short)0, acc01, false, false);
        acc10 = __builtin_amdgcn_wmma_f32_16x16x4_f32(false, a1, false, b0,
                                                      (short)0, acc10, false, false);
        acc11 = __builtin_amdgcn_wmma_f32_16x16x4_f32(false, a1, false, b1,
                                                      (short)0, acc11, false, false);
    }

    const int bb = p >> 14;          // batch
    const int q  = p & (HWo - 1);    // ho*WO+wo
    float* __restrict__ gout = out;
    float* __restrict__ cout = out + Bn * On * HWo;   // 2097152

#pragma unroll
    for (int v = 0; v < 8; ++v) {
        const int m = v + (half << 3);
        {   // O-tile 0: o = m
            const float nom2 = acc00[v], den2 = acc01[v];
            const float g  = (nom2 / (den2 + EPSF) + bias[m]) * 2.0f;  // *STRIDE
            const float cg = den2 / (S_cw + EPSF) * 0.25f;             // /STRIDE^2
            const int oi = ((bb << 5) + m) * HWo + q;
            gout[oi] = g;
            cout[oi] = cg;
        }
        {   // O-tile 1: o = 16 + m
            const int o = 16 + m;
            const float nom2 = acc10[v], den2 = acc11[v];
            const float g  = (nom2 / (den2 + EPSF) + bias[o]) * 2.0f;
            const float cg = den2 / (S_cw + EPSF) * 0.25f;
            const int oi = ((bb << 5) + o) * HWo + q;
            gout[oi] = g;
            cout[oi] = cg;
        }
    }
}

// ---------------------------------------------------------------------------
// launch
// ---------------------------------------------------------------------------
extern "C" void kernel_launch(void* const* d_in, const int* in_sizes, int n_in,
                              void* d_out, int out_size, void* d_ws, size_t ws_size,
                              hipStream_t stream) {
    (void)in_sizes; (void)n_in; (void)out_size; (void)ws_size;

    const float* d     = (const float*)d_in[0];
    const float* cd    = (const float*)d_in[1];
    const float* s     = (const float*)d_in[2];
    /* d_in[3] = cs : unused by the reference */
    const float* gx    = (const float*)d_in[4];
    const float* cgx   = (const float*)d_in[5];
    const float* sprod = (const float*)d_in[6];
    const float* wprop = (const float*)d_in[7];
    const float* spw   = (const float*)d_in[8];
    const float* chw   = (const float*)d_in[9];
    const float* bias  = (const float*)d_in[10];

    float* ws  = (float*)d_ws;
    float* wp  = ws;            // 32
    float* swp = ws + 32;       // 288
    float* cwp = ws + 320;      // 1024
    float* ssw = ws + 1344;     // 1
    float* scw = ws + 1345;     // 1
    float* t0  = ws + 4096;     // Cn*Pn = 2097152 floats (8MB)
    float* t1  = t0 + (size_t)Cn * Pn;
    float* out = (float*)d_out;

    setup_kernel<<<1, 256, 0, stream>>>(wprop, spw, chw, wp, swp, cwp, ssw, scw);

    fused_spatial_kernel<<<(Bn * Cn * HOn * WOn) / 256, 256, 0, stream>>>(
        d, cd, s, gx, cgx, sprod, wp, swp, ssw, t0, t1);

    channel_wmma_kernel<<<(Pn / 16) * 32 / 256, 256, 0, stream>>>(
        t0, t1, cwp, scw, bias, out);
}